// KDEMixture_88003879895535
// MI455X (gfx1250) — compile-verified
//
#include <hip/hip_runtime.h>
#include <hip/hip_bf16.h>

// KDE mixture on gfx1250 (MI455X, wave32).
//
// out[i] = sum_j w[j] * exp(-||p_i - s_j||^2 / (2 h^2))
//
// 16x16 pair tiles via V_WMMA_F32_16X16X4_F32 with the Gram trick, with the
// exp2 scale factor coef = -log2(e)/(2h^2) folded directly into B and p2:
//   A row m = [px, py, pz, 1]                             (16x4, K pad of D=3)
//   B col n = coef * [-2sx, -2sy, -2sz, s2]               (4x16)
//   C[m][n] = coef*(-2 p.s + s2);  arg = min(C + coef*p2[m], 0);  e = exp2(arg)
// Per pair: add + min (dual-issued) + v_exp_f32 + half a pk_fma. The 2e8
// v_exp_f32 are the algorithmic floor. Inputs <0.5MB -> L2-resident.
// Loop is software-pipelined (next tile's loads issued before current math)
// and wave-uniform (readfirstlane) so it lowers to a scalar branch with
// EXEC all-ones throughout -- required by WMMA and free of per-iteration
// exec-mask bookkeeping.

typedef __attribute__((ext_vector_type(2))) float v2f;
typedef __attribute__((ext_vector_type(8))) float v8f;

__global__ __launch_bounds__(256)
void kde_wmma_kernel(const float* __restrict__ points,
                     const float* __restrict__ spatial,
                     const float* __restrict__ weights,
                     const float* __restrict__ bandwidth,
                     float* __restrict__ out,
                     int M, int N) {
    const int tid  = threadIdx.x;
    const int lane = tid & 31;
    const int col  = lane & 15;         // tile row/col index held by this lane
    const int half = lane >> 4;         // 0: K={0,1}, 1: K={2,3} (A/B layout)
    const int wave = tid >> 5;          // 0..7
    // Wave-uniform copy: makes the tile loop bounds scalar (s_cmp/s_cbranch,
    // no per-iteration v_cmp + exec masking; EXEC provably all-ones at WMMA).
    const int waveU = __builtin_amdgcn_readfirstlane(wave);

    // ---- A tile: 16 points, loaded once per block -------------------------
    const int m0 = blockIdx.x * 16;
    int mi = m0 + col;
    mi = mi < M ? mi : (M - 1);         // clamp; extra rows never stored
    const float px = points[3 * mi + 0];
    const float py = points[3 * mi + 1];
    const float pz = points[3 * mi + 2];
    const float p2 = fmaf(px, px, fmaf(py, py, pz * pz));

    v2f a;
    a.x = half ? pz   : px;             // lanes 0-15: K=0 ; lanes 16-31: K=2
    a.y = half ? 1.0f : py;             // lanes 0-15: K=1 ; lanes 16-31: K=3

    const float bw   = bandwidth[0];
    const float coef = -1.4426950408889634f / (2.0f * bw * bw); // -log2e/(2h^2) < 0
    const float k2   = -2.0f * coef;                            // > 0

    // coef * p2 per C-layout row: VGPR v holds row (v + 8*half); row r in lane r.
    float p2c[8];
#pragma unroll
    for (int v = 0; v < 8; ++v)
        p2c[v] = coef * __shfl(p2, v + 8 * half, 32);

    const v8f czero = {};
    float acc[8] = {0.f, 0.f, 0.f, 0.f, 0.f, 0.f, 0.f, 0.f};

    // ---- software-pipelined sweep over column tiles -----------------------
    // Wave w handles tiles w, w+8, w+16, ...
    const int NT = (N + 15) >> 4;       // 625 for N=10000

    float sx = 0.f, sy = 0.f, sz = 0.f, wv = 0.f;
    {   // prologue: load first tile (uniform condition)
        int t0 = waveU < NT ? waveU : 0;
        int n  = t0 * 16 + col;
        int nc = n < N ? n : (N - 1);
        sx = spatial[3 * nc + 0];
        sy = spatial[3 * nc + 1];
        sz = spatial[3 * nc + 2];
        float wr = weights[nc];         // unconditional load ...
        wv = (n < N) ? wr : 0.0f;       // ... then branchless select
    }

    for (int t = waveU; t < NT; t += 8) {
        // current tile operands
        const float csx = sx, csy = sy, csz = sz, cw = wv;

        // issue next tile's loads now; the wait lands after this tile's math
        {
            int tn = (t + 8 < NT) ? (t + 8) : t;    // uniform clamp: harmless reload
            int n  = tn * 16 + col;
            int nc = n < N ? n : (N - 1);
            // deep prefetch two tiles ahead (temporal; data is L2-resident)
            __builtin_prefetch(&spatial[3 * (nc + 16 * 8)], 0, 3);
            sx = spatial[3 * nc + 0];
            sy = spatial[3 * nc + 1];
            sz = spatial[3 * nc + 2];
            float wr = weights[nc];
            wv = (n < N) ? wr : 0.0f;
        }

        // B column, pre-scaled by coef
        const float s2 = fmaf(csx, csx, fmaf(csy, csy, csz * csz));
        v2f b;
        b.x = half ? (k2 * csz)   : (k2 * csx);     // K=2 / K=0
        b.y = half ? (coef * s2)  : (k2 * csy);     // K=3 / K=1

        // C = coef * (-2 p.s + s2)   (EXEC is all-ones here)
        v8f c = __builtin_amdgcn_wmma_f32_16x16x4_f32(
            /*neg_a=*/false, a, /*neg_b=*/false, b,
            /*c_mod=*/(short)0, czero, /*reuse_a=*/false, /*reuse_b=*/false);

#pragma unroll
        for (int v = 0; v < 8; ++v) {
            // coef<0: max(sqdist,0)*coef == min(coef*sqdist, 0)
            const float arg = fminf(c[v] + p2c[v], 0.0f);
            acc[v] = fmaf(__builtin_amdgcn_exp2f(arg), cw, acc[v]);
        }
    }

    // ---- deterministic reduction ------------------------------------------
    // acc[v]: lanes 0-15 hold row v, lanes 16-31 hold row v+8 (cols = lane%16).
    // Butterfly within each 16-lane half -> full per-row sum in every lane.
#pragma unroll
    for (int v = 0; v < 8; ++v) {
        float s = acc[v];
        s += __shfl_xor(s, 1, 16);
        s += __shfl_xor(s, 2, 16);
        s += __shfl_xor(s, 4, 16);
        s += __shfl_xor(s, 8, 16);
        acc[v] = s;
    }

    __shared__ float partial[8][16];
    if (lane == 0) {
#pragma unroll
        for (int v = 0; v < 8; ++v) partial[wave][v] = acc[v];
    } else if (lane == 16) {
#pragma unroll
        for (int v = 0; v < 8; ++v) partial[wave][8 + v] = acc[v];
    }
    __syncthreads();

    if (tid < 16) {
        float s = 0.0f;
#pragma unroll
        for (int w = 0; w < 8; ++w) s += partial[w][tid];   // fixed order
        const int m = m0 + tid;
        if (m < M) out[m] = s;
    }
}

extern "C" void kernel_launch(void* const* d_in, const int* in_sizes, int n_in,
                              void* d_out, int out_size, void* d_ws, size_t ws_size,
                              hipStream_t stream) {
    const float* points    = (const float*)d_in[0];
    const float* spatial   = (const float*)d_in[1];
    const float* weights   = (const float*)d_in[2];
    const float* bandwidth = (const float*)d_in[3];
    float* out = (float*)d_out;

    const int M = in_sizes[0] / 3;   // 20000
    const int N = in_sizes[1] / 3;   // 10000

    const int grid = (M + 15) / 16;  // 1250 blocks x 8 wave32
    kde_wmma_kernel<<<grid, 256, 0, stream>>>(points, spatial, weights, bandwidth,
                                              out, M, N);
}